// Decoder_20014547599839
// MI455X (gfx1250) — compile-verified
//
#include <hip/hip_runtime.h>
#include <hip/hip_bf16.h>
#include <cstdint>

// Problem constants (from reference): B=16, S=16, H=501, C=10, LAT=56, G=H+S
#define Bv 16
#define Sv 16
#define Hv 501
#define Cv 10
#define LATv 56
#define Gv 517
#define H2v 1002
#define H3v 1503
#define H4v 2004
// Padded leading dims (K inputs must be multiples of 4, padding zero-filled)
#define HP 504
#define H2P 1004
#define H3P 1504
#define CP 12

#define NTHREADS 1024
#define NWAVES 32

typedef __attribute__((ext_vector_type(2))) float v2f;
typedef __attribute__((ext_vector_type(8))) float v8f;

__device__ __forceinline__ float sigm(float x) { return 1.0f / (1.0f + __expf(-x)); }

__device__ __forceinline__ int imin(int a, int b) { return a < b ? a : b; }

// Device-scope fence + workgroup barrier: a 1024-thread workgroup spans both
// CUs of the WGP, and scratch lives in global memory, so release/acquire must
// reach past each CU's L0.
__device__ __forceinline__ void barsync() {
    __threadfence();
    __syncthreads();
}

// out(16 x N) = A(16 x K) @ W(N x K)^T + bias, optional ReLU.
// A: row-major, lda multiple of 4, >= K, padding zero-filled -> one b64 load.
// W: row-major (N x K) exactly as stored in d_in.
// All out-of-range accesses are handled with BRANCH-FREE index clamps:
//  - kb >= K   -> A element is 0 (padding), so clamped W value is multiplied by 0
//  - col >= N  -> result column discarded at the (guarded) store
// Uses V_WMMA_F32_16X16X4_F32; M=16 matches the batch exactly.
__device__ __forceinline__ void gemm16(
    const float* __restrict__ A, int lda,
    const float* __restrict__ W, int ldw,
    const float* __restrict__ bias, int bstride,
    float* __restrict__ out, int ldo,
    int N, int K, bool relu)
{
    const int tid  = threadIdx.x;
    const int wave = tid >> 5;       // wave32
    const int lane = tid & 31;
    const int hi   = lane >> 4;      // 0: lanes 0-15, 1: lanes 16-31
    const int l16  = lane & 15;
    const int ntiles = (N + 15) >> 4;

    // A 16x4 fragment: VGPR0 = K {0,2}, VGPR1 = K {1,3}; row = lane&15.
    // Base for this lane's row; kb = kk + 2*hi is even -> 8B-aligned v2f load.
    const float* arow = A + (size_t)l16 * lda + (hi << 1);

    for (int t = wave; t < ntiles; t += NWAVES) {
        const int n0   = t << 4;
        const int col  = n0 + l16;
        const bool colok = col < N;
        const int colc = colok ? col : (N - 1);
        const float* wrow = W + (size_t)colc * (size_t)ldw;

        v8f c = {0.f, 0.f, 0.f, 0.f, 0.f, 0.f, 0.f, 0.f};
#pragma unroll 2
        for (int kk = 0; kk < K; kk += 4) {
            const int kb = kk + (hi << 1);
            v2f a = *(const v2f*)(arow + kk);
            v2f b;
            b.x = wrow[imin(kb,     K - 1)];
            b.y = wrow[imin(kb + 1, K - 1)];
            c = __builtin_amdgcn_wmma_f32_16x16x4_f32(false, a, false, b,
                                                      (short)0, c, false, false);
        }
        if (colok) {
            const float bval = bias ? bias[(size_t)colc * (size_t)bstride] : 0.f;
#pragma unroll
            for (int r = 0; r < 8; ++r) {
                const int m = r + (hi << 3);   // C/D: VGPR r -> M=r (lo) / r+8 (hi)
                float v = c[r] + bval;
                if (relu) v = fmaxf(v, 0.f);
                out[(size_t)m * ldo + col] = v;
            }
        }
    }
}

// hv = (1-u)*n + u*hprev from precomputed gi (x-path) and gh (h-path) pre-activations.
__device__ __forceinline__ void gru_combine(int tid,
    const float* __restrict__ gi, const float* __restrict__ gh,
    const float* __restrict__ hprev, float* __restrict__ hv,
    float* __restrict__ nrow)
{
    for (int e = tid; e < Bv * Hv; e += NTHREADS) {
        const int b = e / Hv, h = e - b * Hv;
        const float* gib = gi + (size_t)b * H3P;
        const float* ghb = gh + (size_t)b * H3P;
        const float r = sigm(gib[h] + ghb[h]);
        const float u = sigm(gib[Hv + h] + ghb[Hv + h]);
        const float n = tanhf(gib[2 * Hv + h] + r * ghb[2 * Hv + h]);
        const float v = (1.f - u) * n + u * hprev[(size_t)b * HP + h];
        hv[(size_t)b * HP + h] = v;
        if (nrow) nrow[(size_t)b * HP + h] = v;
    }
}

__global__ __launch_bounds__(NTHREADS)
void decoder_gfx1250_kernel(
    const float* __restrict__ z, const float* __restrict__ dep,
    const float* __restrict__ nodes,
    const float* __restrict__ lin1_w, const float* __restrict__ lin1_b,
    const float* __restrict__ av1_w,  const float* __restrict__ av1_b,
    const float* __restrict__ av2_w,  const float* __restrict__ av2_b,
    const float* __restrict__ ae1_w,  const float* __restrict__ ae1_b,
    const float* __restrict__ ae2_w,  const float* __restrict__ ae2_b,
    const float* __restrict__ gate_w, const float* __restrict__ gate_b,
    const float* __restrict__ map_w,  const float* __restrict__ map_b,
    const float* __restrict__ wih,    const float* __restrict__ bih,
    const float* __restrict__ whh,    const float* __restrict__ bhh,
    float* __restrict__ out, float* __restrict__ ws)
{
    const int tid = threadIdx.x;

    // Hot recurrent state in LDS (~64 KB of the 320 KB WGP pool)
    __shared__ float sHV [Bv * HP];   // current hidden hv
    __shared__ float sHIN[Bv * HP];   // h_in (masked gated sum)
    __shared__ float sXT [Bv * CP];   // x_t padded to 12 cols

    // Global scratch layout (floats); total ~515520 floats (~2.06 MB)
    float* NHS  = ws;                       // [S][B][HP] node hidden states
    float* QG   = NHS  + Sv * Bv * HP;      // [S][B][HP] cached gate pre-acts
    float* QM   = QG   + Sv * Bv * HP;      // [S][B][HP] cached map  pre-acts
    float* H0b  = QM   + Sv * Bv * HP;      // [B][HP] initial GRU hidden
    float* GST  = H0b  + Bv * HP;           // [B][HP] graph_state
    float* GI   = GST  + Bv * HP;           // [B][H3P] x-path GRU pre-acts
    float* GH   = GI   + Bv * H3P;          // [B][H3P] h-path GRU pre-acts
    float* CAT  = GH   + Bv * H3P;          // [B][H2P] concat(hv, nhs[vj])
    float* EHID = CAT  + Bv * H2P;          // [B][H4v] add_edge hidden (2004 % 4 == 0)
    float* AVH  = EHID + Bv * H4v;          // [B][H2P] add_vertex hidden
    const int WS_TOT = 3 * Sv * Bv * HP + 2 * Bv * HP + 2 * Bv * H3P
                     + Bv * H2P + Bv * H4v + Bv * H2P;

    // ---- deterministic re-init every call (harness does not re-poison) ----
    for (int i = tid; i < WS_TOT; i += NTHREADS) ws[i] = 0.f;
    for (int i = tid; i < Bv * Sv * Sv + Bv * Sv * Cv; i += NTHREADS) out[i] = 0.f;
    for (int i = tid; i < Bv * HP; i += NTHREADS) { sHV[i] = 0.f; sHIN[i] = 0.f; }
    for (int i = tid; i < Bv * CP; i += NTHREADS) sXT[i] = 0.f;
    barsync();

    // q caches start at the identity-column contribution (nhs == 0 everywhere)
    for (int e = tid; e < Sv * Bv * Hv; e += NTHREADS) {
        const int s = e / (Bv * Hv);
        const int r = e - s * Bv * Hv;
        const int b = r / Hv, h = r - b * Hv;
        QG[((size_t)s * Bv + b) * HP + h] = gate_w[(size_t)h * Gv + Hv + s];
        QM[((size_t)s * Bv + b) * HP + h] = map_w [(size_t)h * Gv + Hv + s];
    }
    // Warm L2 (weights ~16 MB, L2 = 192 MB): global_prefetch_b8
    for (int i = tid * 16; i < H4v * H2v; i += NTHREADS * 16)
        __builtin_prefetch(ae1_w + i, 0, 1);
    for (int i = tid * 16; i < Hv * Gv; i += NTHREADS * 16) {
        __builtin_prefetch(gate_w + i, 0, 1);
        __builtin_prefetch(map_w  + i, 0, 1);
    }
    for (int i = tid * 16; i < 3 * Hv * Hv; i += NTHREADS * 16)
        __builtin_prefetch(whh + i, 0, 1);
    barsync();

    // graph_state = z @ lin1_w^T + lin1_b          (K=56 is already a mult of 4)
    gemm16(z, LATv, lin1_w, LATv, lin1_b, 1, GST, HP, Hv, LATv, false);
    barsync();

    for (int index = 0; index < Sv; ++index) {
        // stage x_t and h0
        for (int i = tid; i < Bv * Cv; i += NTHREADS) {
            const int b = i / Cv, c = i - b * Cv;
            sXT[b * CP + c] = nodes[((size_t)b * Sv + index) * Cv + c];
        }
        for (int i = tid; i < Bv * HP; i += NTHREADS)
            H0b[i] = (index == 0) ? GST[i] : 0.f;
        barsync();

        // gen_nodes[index] = add_vertex(graph_state)
        gemm16(GST, HP, av1_w, Hv, av1_b, 1, AVH, H2P, H2v, Hv, true);
        barsync();
        gemm16(AVH, H2P, av2_w, H2v, av2_b, 1,
               out + Bv * Sv * Sv + index * Cv, Sv * Cv, Cv, H2v, false);
        // gi = x_t @ W_ih^T + b_ih (constant over the inner scan)
        gemm16(sXT, CP, wih, Cv, bih, 1, GI, H3P, H3v, Cv, false);
        barsync();

        // hv = gru(x_t, h0)
        gemm16(H0b, HP, whh, Hv, bhh, 1, GH, H3P, H3v, Hv, false);
        barsync();
        gru_combine(tid, GI, GH, H0b, sHV, nullptr);  // nhs[:,index] stays 0
        barsync();

        for (int vj = index - 1; vj >= 0; --vj) {
            // edge = add_edge(concat(hv, nhs[:,vj]))
            const float* nrow = NHS + (size_t)vj * Bv * HP;
            for (int i = tid; i < Bv * Hv; i += NTHREADS) {
                const int b = i / Hv, h = i - b * Hv;
                CAT[(size_t)b * H2P + h]      = sHV[(size_t)b * HP + h];
                CAT[(size_t)b * H2P + Hv + h] = nrow[(size_t)b * HP + h];
            }
            barsync();
            gemm16(CAT, H2P, ae1_w, H2v, ae1_b, 1, EHID, H4v, H4v, H2v, true);
            // refresh q caches for row `index` from current nhs[:,index]
            const float* nidx = NHS + (size_t)index * Bv * HP;
            gemm16(nidx, HP, gate_w, Gv, gate_w + Hv + index, Gv,
                   QG + (size_t)index * Bv * HP, HP, Hv, Hv, false);
            gemm16(nidx, HP, map_w, Gv, map_w + Hv + index, Gv,
                   QM + (size_t)index * Bv * HP, HP, Hv, Hv, false);
            barsync();
            // gen_dep[:, index, vj] = edge
            gemm16(EHID, H4v, ae2_w, H4v, ae2_b, 1,
                   out + index * Sv + vj, Sv * Sv, 1, H4v, false);

            // h_in[b,h] = sum_s am * sigmoid(am*qg + gb) * (am*qm + mb)
            for (int e = tid; e < Bv * Hv; e += NTHREADS) {
                const int b = e / Hv, h = e - b * Hv;
                const float gb = gate_b[h], mb = map_b[h];
                const float* anc = dep + ((size_t)b * Sv + index) * Sv;
                float acc = 0.f;
                for (int s = vj; s < Sv; ++s) {
                    const float am  = anc[s];
                    const float qgv = QG[((size_t)s * Bv + b) * HP + h];
                    const float qmv = QM[((size_t)s * Bv + b) * HP + h];
                    acc += am * sigm(am * qgv + gb) * (am * qmv + mb);
                }
                sHIN[(size_t)b * HP + h] = acc;
            }
            barsync();
            // hv_new = gru(x_t, h_in); nhs[:,index] = hv_new
            gemm16(sHIN, HP, whh, Hv, bhh, 1, GH, H3P, H3v, Hv, false);
            barsync();
            gru_combine(tid, GI, GH, sHIN, sHV, NHS + (size_t)index * Bv * HP);
            barsync();
        }

        // finalize: nhs[:,index] = hv; graph_state = hv
        for (int i = tid; i < Bv * HP; i += NTHREADS) {
            const float v = sHV[i];
            NHS[(size_t)index * Bv * HP + i] = v;
            GST[i] = v;
        }
        barsync();
        // refresh q caches for the now-final row `index`
        const float* nfin = NHS + (size_t)index * Bv * HP;
        gemm16(nfin, HP, gate_w, Gv, gate_w + Hv + index, Gv,
               QG + (size_t)index * Bv * HP, HP, Hv, Hv, false);
        gemm16(nfin, HP, map_w, Gv, map_w + Hv + index, Gv,
               QM + (size_t)index * Bv * HP, HP, Hv, Hv, false);
        barsync();
    }
}

extern "C" void kernel_launch(void* const* d_in, const int* in_sizes, int n_in,
                              void* d_out, int out_size, void* d_ws, size_t ws_size,
                              hipStream_t stream)
{
    (void)in_sizes; (void)n_in; (void)out_size; (void)ws_size; // ws need ~2.1 MB
    const float* z      = (const float*)d_in[0];
    const float* dep    = (const float*)d_in[1];
    const float* nodes  = (const float*)d_in[2];
    const float* lin1_w = (const float*)d_in[3];
    const float* lin1_b = (const float*)d_in[4];
    const float* av1_w  = (const float*)d_in[5];
    const float* av1_b  = (const float*)d_in[6];
    const float* av2_w  = (const float*)d_in[7];
    const float* av2_b  = (const float*)d_in[8];
    const float* ae1_w  = (const float*)d_in[9];
    const float* ae1_b  = (const float*)d_in[10];
    const float* ae2_w  = (const float*)d_in[11];
    const float* ae2_b  = (const float*)d_in[12];
    const float* gate_w = (const float*)d_in[13];
    const float* gate_b = (const float*)d_in[14];
    const float* map_w  = (const float*)d_in[15];
    const float* map_b  = (const float*)d_in[16];
    const float* wih    = (const float*)d_in[17];
    const float* bih    = (const float*)d_in[18];
    const float* whh    = (const float*)d_in[19];
    const float* bhh    = (const float*)d_in[20];

    // One persistent workgroup (32 wave32 waves on one WGP) runs the whole
    // sequential recurrence; the problem is latency-bound, not throughput-bound.
    decoder_gfx1250_kernel<<<1, NTHREADS, 0, stream>>>(
        z, dep, nodes, lin1_w, lin1_b, av1_w, av1_b, av2_w, av2_b,
        ae1_w, ae1_b, ae2_w, ae2_b, gate_w, gate_b, map_w, map_b,
        wih, bih, whh, bhh, (float*)d_out, (float*)d_ws);
}